// GAT_44006234914926
// MI455X (gfx1250) — compile-verified
//
#include <hip/hip_runtime.h>
#include <hip/hip_bf16.h>

typedef __attribute__((ext_vector_type(2))) float v2f;
typedef __attribute__((ext_vector_type(8))) float v8f;

namespace {
constexpr int kN   = 100000;    // nodes
constexpr int kE   = 1600000;   // edges (without self loops)
constexpr int kEA  = kE + kN;   // edges + self loops
constexpr int kF   = 256;       // input features
constexpr int kH1  = 8;         // layer-1 heads
constexpr int kHid = 8;         // layer-1 channels per head
constexpr int kC1  = kH1 * kHid;// 64
constexpr int kC2  = 40;        // layer-2 classes (1 head)
constexpr float kSlope = 0.2f;
}

// ---------------------------------------------------------------- utilities
__global__ void fill_f32(float* __restrict__ p, float v, int n) {
  int i = blockIdx.x * blockDim.x + threadIdx.x;
  if (i < n) p[i] = v;
}

__device__ __forceinline__ void atomicMaxF32(float* addr, float val) {
  // classic monotonic-bit-pattern trick; init value must be -inf
  if (val >= 0.0f) atomicMax((int*)addr, __float_as_int(val));
  else             atomicMin((unsigned int*)addr, __float_as_uint(val));
}

__device__ __forceinline__ float leaky(float v) {
  return v > 0.0f ? v : kSlope * v;
}

// --------------------------------------------------- GEMM1: h = x@W1  (f32 WMMA)
// grid.x = kN/16 row tiles, 128 threads = 4 waves, wave w covers cols [16w,16w+16)
__global__ __launch_bounds__(128) void gemm1_wmma(const float* __restrict__ x,
                                                  const float* __restrict__ W,
                                                  float* __restrict__ h) {
  const int row0 = blockIdx.x * 16;
  const int wave = threadIdx.x >> 5;
  const int lane = threadIdx.x & 31;
  const int col0 = wave * 16;
  const int m    = lane & 15;          // A row this lane holds
  const int half = lane >> 4;          // 0: K pair {0,1}, 1: K pair {2,3}
  const int kh   = half * 2;
  const int nc   = lane & 15;          // B/D column this lane holds

  v8f acc = {};
  const float* xrow = x + (size_t)(row0 + m) * kF;
  #pragma unroll 4
  for (int k = 0; k < kF; k += 4) {
    v2f a, b;
    a.x = xrow[k + kh];
    a.y = xrow[k + kh + 1];
    b.x = W[(size_t)(k + kh)     * kC1 + col0 + nc];
    b.y = W[(size_t)(k + kh + 1) * kC1 + col0 + nc];
    acc = __builtin_amdgcn_wmma_f32_16x16x4_f32(false, a, false, b,
                                                (short)0, acc, false, false);
  }
  // C/D layout: vgpr r holds row (r + 8*half), col nc
  #pragma unroll
  for (int r = 0; r < 8; ++r)
    h[(size_t)(row0 + r + 8 * half) * kC1 + col0 + nc] = acc[r];
}

// --------------------------------------------------- GEMM2: g = h2@W2 (f32 WMMA)
// grid.x = kN/16, 96 threads = 3 waves covering 48 cols; cols >= 40 masked.
__global__ __launch_bounds__(96) void gemm2_wmma(const float* __restrict__ h2,
                                                 const float* __restrict__ W,
                                                 float* __restrict__ g) {
  const int row0 = blockIdx.x * 16;
  const int wave = threadIdx.x / 32;
  const int lane = threadIdx.x & 31;
  const int col0 = wave * 16;
  const int m    = lane & 15;
  const int half = lane >> 4;
  const int kh   = half * 2;
  const int col  = col0 + (lane & 15);
  const bool valid = col < kC2;

  v8f acc = {};
  const float* hrow = h2 + (size_t)(row0 + m) * kC1;
  #pragma unroll
  for (int k = 0; k < kC1; k += 4) {
    v2f a, b;
    a.x = hrow[k + kh];
    a.y = hrow[k + kh + 1];
    b.x = valid ? W[(size_t)(k + kh)     * kC2 + col] : 0.0f;  // select, no branch
    b.y = valid ? W[(size_t)(k + kh + 1) * kC2 + col] : 0.0f;
    acc = __builtin_amdgcn_wmma_f32_16x16x4_f32(false, a, false, b,
                                                (short)0, acc, false, false);
  }
  if (valid) {
    #pragma unroll
    for (int r = 0; r < 8; ++r)
      g[(size_t)(row0 + r + 8 * half) * kC2 + col] = acc[r];
  }
}

// -------------------------------------------- per-node attention coefficients
__global__ void attn1(const float* __restrict__ h,
                      const float* __restrict__ a_src, const float* __restrict__ a_dst,
                      float* __restrict__ as, float* __restrict__ ad) {
  int idx = blockIdx.x * blockDim.x + threadIdx.x;     // node*8 + head
  if (idx >= kN * kH1) return;
  int node = idx >> 3, head = idx & 7;
  const float* hp = h + (size_t)node * kC1 + head * kHid;
  float s = 0.f, d = 0.f;
  #pragma unroll
  for (int c = 0; c < kHid; ++c) {
    s += hp[c] * a_src[head * kHid + c];
    d += hp[c] * a_dst[head * kHid + c];
  }
  as[idx] = s; ad[idx] = d;
}

__global__ void attn2(const float* __restrict__ g,
                      const float* __restrict__ a_src, const float* __restrict__ a_dst,
                      float* __restrict__ as, float* __restrict__ ad) {
  int node = blockIdx.x * blockDim.x + threadIdx.x;
  if (node >= kN) return;
  const float* gp = g + (size_t)node * kC2;
  float s = 0.f, d = 0.f;
  for (int c = 0; c < kC2; ++c) { s += gp[c] * a_src[c]; d += gp[c] * a_dst[c]; }
  as[node] = s; ad[node] = d;
}

// -------------------------------------------- edge softmax passes, layer 1 (8 heads)
__device__ __forceinline__ void edge_sd(int e, const int* srcI, const int* dstI,
                                        int& s, int& d) {
  s = (e < kE) ? srcI[e] : (e - kE);
  d = (e < kE) ? dstI[e] : (e - kE);
}

__global__ void edge_max1(const int* __restrict__ srcI, const int* __restrict__ dstI,
                          const float* __restrict__ as, const float* __restrict__ ad,
                          float* __restrict__ amax) {
  int e = blockIdx.x * blockDim.x + threadIdx.x;
  if (e >= kEA) return;
  int s, d; edge_sd(e, srcI, dstI, s, d);
  #pragma unroll
  for (int hh = 0; hh < kH1; ++hh)
    atomicMaxF32(&amax[d * kH1 + hh], leaky(as[s * kH1 + hh] + ad[d * kH1 + hh]));
}

__global__ void edge_sum1(const int* __restrict__ srcI, const int* __restrict__ dstI,
                          const float* __restrict__ as, const float* __restrict__ ad,
                          const float* __restrict__ amax, float* __restrict__ den) {
  int e = blockIdx.x * blockDim.x + threadIdx.x;
  if (e >= kEA) return;
  int s, d; edge_sd(e, srcI, dstI, s, d);
  #pragma unroll
  for (int hh = 0; hh < kH1; ++hh) {
    float v = leaky(as[s * kH1 + hh] + ad[d * kH1 + hh]);
    atomicAdd(&den[d * kH1 + hh], __expf(v - amax[d * kH1 + hh]));
  }
}

// one thread per (edge, head): 8 consecutive-channel atomic adds
__global__ void edge_aggr1(const int* __restrict__ srcI, const int* __restrict__ dstI,
                           const float* __restrict__ as, const float* __restrict__ ad,
                           const float* __restrict__ amax, const float* __restrict__ den,
                           const float* __restrict__ h, float* __restrict__ out) {
  long long idx = (long long)blockIdx.x * blockDim.x + threadIdx.x;
  if (idx >= (long long)kEA * kH1) return;
  int e = (int)(idx >> 3), hh = (int)(idx & 7);
  int s, d; edge_sd(e, srcI, dstI, s, d);
  float v = leaky(as[s * kH1 + hh] + ad[d * kH1 + hh]);
  float alpha = __expf(v - amax[d * kH1 + hh]) / (den[d * kH1 + hh] + 1e-16f);
  const float* hp = h + (size_t)s * kC1 + hh * kHid;
  float* op = out + (size_t)d * kC1 + hh * kHid;
  #pragma unroll
  for (int c = 0; c < kHid; ++c) atomicAdd(&op[c], hp[c] * alpha);
}

// -------------------------------------------- elu(out1 + b1) -> h2
__global__ void elu1(const float* __restrict__ out1, const float* __restrict__ b,
                     float* __restrict__ h2) {
  int i = blockIdx.x * blockDim.x + threadIdx.x;
  if (i >= kN * kC1) return;
  float v = out1[i] + b[i % kC1];
  h2[i] = v > 0.0f ? v : (__expf(v) - 1.0f);
}

// -------------------------------------------- edge softmax passes, layer 2 (1 head)
__global__ void edge_max2(const int* __restrict__ srcI, const int* __restrict__ dstI,
                          const float* __restrict__ as, const float* __restrict__ ad,
                          float* __restrict__ amax) {
  int e = blockIdx.x * blockDim.x + threadIdx.x;
  if (e >= kEA) return;
  int s, d; edge_sd(e, srcI, dstI, s, d);
  atomicMaxF32(&amax[d], leaky(as[s] + ad[d]));
}

__global__ void edge_sum2(const int* __restrict__ srcI, const int* __restrict__ dstI,
                          const float* __restrict__ as, const float* __restrict__ ad,
                          const float* __restrict__ amax, float* __restrict__ den) {
  int e = blockIdx.x * blockDim.x + threadIdx.x;
  if (e >= kEA) return;
  int s, d; edge_sd(e, srcI, dstI, s, d);
  atomicAdd(&den[d], __expf(leaky(as[s] + ad[d]) - amax[d]));
}

__global__ void edge_aggr2(const int* __restrict__ srcI, const int* __restrict__ dstI,
                           const float* __restrict__ as, const float* __restrict__ ad,
                           const float* __restrict__ amax, const float* __restrict__ den,
                           const float* __restrict__ g, float* __restrict__ out) {
  int e = blockIdx.x * blockDim.x + threadIdx.x;
  if (e >= kEA) return;
  int s, d; edge_sd(e, srcI, dstI, s, d);
  float alpha = __expf(leaky(as[s] + ad[d]) - amax[d]) / (den[d] + 1e-16f);
  const float* gp = g + (size_t)s * kC2;
  float* op = out + (size_t)d * kC2;
  for (int c = 0; c < kC2; ++c) atomicAdd(&op[c], gp[c] * alpha);
}

__global__ void bias2(float* __restrict__ out, const float* __restrict__ b) {
  int i = blockIdx.x * blockDim.x + threadIdx.x;
  if (i >= kN * kC2) return;
  out[i] += b[i % kC2];
}

// ------------------------------------------------------------------ launcher
extern "C" void kernel_launch(void* const* d_in, const int* in_sizes, int n_in,
                              void* d_out, int out_size, void* d_ws, size_t ws_size,
                              hipStream_t stream) {
  (void)in_sizes; (void)n_in; (void)out_size; (void)ws_size;
  const float* x        = (const float*)d_in[0];
  const int*   srcI     = (const int*)d_in[1];          // edge_index[0]
  const int*   dstI     = (const int*)d_in[1] + kE;     // edge_index[1]
  const float* W1       = (const float*)d_in[2];
  const float* att_src1 = (const float*)d_in[3];
  const float* att_dst1 = (const float*)d_in[4];
  const float* b1       = (const float*)d_in[5];
  const float* W2       = (const float*)d_in[6];
  const float* att_src2 = (const float*)d_in[7];
  const float* att_dst2 = (const float*)d_in[8];
  const float* b2       = (const float*)d_in[9];
  float* out = (float*)d_out;

  // workspace carve-up (~82 MB of f32); per-node tables fit in the 192MB L2
  float* w = (float*)d_ws;
  float* h1    = w; w += (size_t)kN * kC1;   // 6.4M  (reused as h2 after elu)
  float* as1   = w; w += (size_t)kN * kH1;
  float* ad1   = w; w += (size_t)kN * kH1;
  float* amax1 = w; w += (size_t)kN * kH1;
  float* den1  = w; w += (size_t)kN * kH1;
  float* out1  = w; w += (size_t)kN * kC1;   // 6.4M
  float* g     = w; w += (size_t)kN * kC2;   // 4.0M
  float* as2   = w; w += kN;
  float* ad2   = w; w += kN;
  float* amax2 = w; w += kN;
  float* den2  = w; w += kN;

  const float negInf = -__builtin_inff();
  auto cdiv = [](long long a, long long b) { return (int)((a + b - 1) / b); };

  // init accumulators (harness poisons buffers; must re-init every call)
  fill_f32<<<cdiv(kN * kH1, 256), 256, 0, stream>>>(amax1, negInf, kN * kH1);
  fill_f32<<<cdiv(kN * kH1, 256), 256, 0, stream>>>(den1, 0.0f, kN * kH1);
  fill_f32<<<cdiv(kN * kC1, 256), 256, 0, stream>>>(out1, 0.0f, kN * kC1);
  fill_f32<<<cdiv(kN, 256), 256, 0, stream>>>(amax2, negInf, kN);
  fill_f32<<<cdiv(kN, 256), 256, 0, stream>>>(den2, 0.0f, kN);
  fill_f32<<<cdiv(kN * kC2, 256), 256, 0, stream>>>(out, 0.0f, kN * kC2);

  // ---- layer 1
  gemm1_wmma<<<kN / 16, 128, 0, stream>>>(x, W1, h1);
  attn1<<<cdiv(kN * kH1, 256), 256, 0, stream>>>(h1, att_src1, att_dst1, as1, ad1);
  edge_max1<<<cdiv(kEA, 256), 256, 0, stream>>>(srcI, dstI, as1, ad1, amax1);
  edge_sum1<<<cdiv(kEA, 256), 256, 0, stream>>>(srcI, dstI, as1, ad1, amax1, den1);
  edge_aggr1<<<cdiv((long long)kEA * kH1, 256), 256, 0, stream>>>(
      srcI, dstI, as1, ad1, amax1, den1, h1, out1);
  elu1<<<cdiv(kN * kC1, 256), 256, 0, stream>>>(out1, b1, h1);  // h2 -> h1 buffer

  // ---- layer 2
  gemm2_wmma<<<kN / 16, 96, 0, stream>>>(h1, W2, g);
  attn2<<<cdiv(kN, 256), 256, 0, stream>>>(g, att_src2, att_dst2, as2, ad2);
  edge_max2<<<cdiv(kEA, 256), 256, 0, stream>>>(srcI, dstI, as2, ad2, amax2);
  edge_sum2<<<cdiv(kEA, 256), 256, 0, stream>>>(srcI, dstI, as2, ad2, amax2, den2);
  edge_aggr2<<<cdiv(kEA, 256), 256, 0, stream>>>(srcI, dstI, as2, ad2, amax2, den2, g, out);
  bias2<<<cdiv(kN * kC2, 256), 256, 0, stream>>>(out, b2);
}